// ETSFormer_47931835023802
// MI455X (gfx1250) — compile-verified
//
#include <hip/hip_runtime.h>
#include <math.h>

#define B_  16
#define N_  2048
#define TF_ 7
#define H_  8
#define T_  96
#define D_  512
#define D4_ 2048
#define BNROWS (B_*N_)

typedef float v2f __attribute__((ext_vector_type(2)));
typedef float v8f __attribute__((ext_vector_type(8)));

#if defined(__has_builtin)
#if __has_builtin(__builtin_amdgcn_global_load_async_to_lds_b128) && \
    __has_builtin(__builtin_amdgcn_s_wait_asynccnt)
#define USE_ASYNC_LDS 1
#endif
#endif
#ifndef USE_ASYNC_LDS
#define USE_ASYNC_LDS 0
#endif

#if USE_ASYNC_LDS
typedef int v4i __attribute__((vector_size(16)));
typedef __attribute__((address_space(1))) v4i as1_v4i;
typedef __attribute__((address_space(3))) v4i as3_v4i;
#endif

// ---------------------------------------------------------------- utilities
__global__ void k_zero(float* __restrict__ p, int n) {
  int i = blockIdx.x * 256 + threadIdx.x;
  if (i < n) p[i] = 0.0f;
}

// in [B, R, C] -> out [B, C, R]
__global__ __launch_bounds__(256)
void k_transpose(const float* __restrict__ in, float* __restrict__ out, int R, int C) {
  __shared__ float t[32][33];
  int b = blockIdx.z;
  int c0 = blockIdx.x * 32, r0 = blockIdx.y * 32;
  const float* ib = in + (size_t)b * R * C;
  float* ob = out + (size_t)b * R * C;
  for (int i = threadIdx.y; i < 32; i += 8)
    t[i][threadIdx.x] = ib[(size_t)(r0 + i) * C + c0 + threadIdx.x];
  __syncthreads();
  for (int i = threadIdx.y; i < 32; i += 8)
    ob[(size_t)(c0 + i) * R + r0 + threadIdx.x] = t[threadIdx.x][i];
}

// ---------------------------------------------------------------- embedding
__global__ __launch_bounds__(256)
void k_embed(const float* __restrict__ x, const float* __restrict__ w,
             const float* __restrict__ bias, float* __restrict__ z) {
  int b = blockIdx.z, n = blockIdx.y;
  int d = blockIdx.x * 256 + threadIdx.x;
  float acc = bias[d];
  const float* wr = w + (size_t)d * (TF_ * 3);
#pragma unroll
  for (int j = 0; j < 3; ++j) {
    int nn = n + j - 1;
    if (nn < 0 || nn >= N_) continue;
    const float* xr = x + ((size_t)b * N_ + nn) * TF_;
#pragma unroll
    for (int c = 0; c < TF_; ++c) acc += xr[c] * wr[c * 3 + j];
  }
  z[((size_t)b * N_ + n) * D_ + d] = acc;
}

// ---------------------------------------------------------------- WMMA GEMM
// C[M,Nc] = act(A[M,K] @ W[Nc,K]^T + bias)
// EPI==1: out2 = r ; C = resid - r  (fused z-update + latent capture)
// Block tile 128x64, 8 waves, 32x32 per wave, double-buffered LDS with
// gfx1250 async global->LDS staging (ASYNCcnt) when available.
#define BM 128
#define BN 64
#define BK 16
#define BKP 20   // row stride (floats): 80B = 16B multiple -> aligned B128 LDS stores,
                 // and 20*i mod 64 hits 16 distinct banks -> conflict-free frag reads

template <int ACT, int EPI>
__global__ __launch_bounds__(256)
void k_gemm(const float* __restrict__ A, const float* __restrict__ W,
            const float* __restrict__ bias, float* __restrict__ C,
            float* __restrict__ out2, const float* __restrict__ resid,
            int M, int Nc, int K) {
  __shared__ __align__(16) float As[2][BM][BKP];
  __shared__ __align__(16) float Ws[2][BN][BKP];
  const int tid  = threadIdx.x;
  const int bm   = blockIdx.y * BM;
  const int bn   = blockIdx.x * BN;
  const int wave = tid >> 5;
  const int lane = tid & 31;
  const int wm = (wave >> 1) * 32;  // 0,32,64,96
  const int wn = (wave & 1) * 32;   // 0,32
  const int l15 = lane & 15;
  const int khalf = (lane < 16) ? 0 : 2;

  v8f acc00 = {0.f,0.f,0.f,0.f,0.f,0.f,0.f,0.f};
  v8f acc01 = {0.f,0.f,0.f,0.f,0.f,0.f,0.f,0.f};
  v8f acc10 = {0.f,0.f,0.f,0.f,0.f,0.f,0.f,0.f};
  v8f acc11 = {0.f,0.f,0.f,0.f,0.f,0.f,0.f,0.f};

  // tile staging: 256 threads; A tile 128x16 (2 float4 rows/thread), W tile 64x16 (1)
  const int lr = tid >> 2;        // 0..63
  const int lc = (tid & 3) * 4;   // 0,4,8,12
  const float* Ag0 = A + (size_t)(bm + lr) * K + lc;
  const float* Ag1 = A + (size_t)(bm + lr + 64) * K + lc;
  const float* Wg  = W + (size_t)(bn + lr) * K + lc;

  const int nk = K / BK;

#if USE_ASYNC_LDS
  // ---- prologue: async-copy tile 0 into buffer 0 ----
  __builtin_amdgcn_global_load_async_to_lds_b128(
      (as1_v4i*)(Ag0), (as3_v4i*)&As[0][lr][lc], 0, 0);
  __builtin_amdgcn_global_load_async_to_lds_b128(
      (as1_v4i*)(Ag1), (as3_v4i*)&As[0][lr + 64][lc], 0, 0);
  __builtin_amdgcn_global_load_async_to_lds_b128(
      (as1_v4i*)(Wg), (as3_v4i*)&Ws[0][lr][lc], 0, 0);
  __builtin_amdgcn_s_wait_asynccnt(0);
  __syncthreads();
#else
  {
    float4 ra0 = *(const float4*)Ag0;
    float4 ra1 = *(const float4*)Ag1;
    float4 rw  = *(const float4*)Wg;
    *(float4*)&As[0][lr][lc]      = ra0;
    *(float4*)&As[0][lr + 64][lc] = ra1;
    *(float4*)&Ws[0][lr][lc]      = rw;
  }
  __syncthreads();
#endif

  for (int k = 0; k < nk; ++k) {
    const int cur = k & 1;
#if USE_ASYNC_LDS
    if (k + 1 < nk) {
      const int ko = (k + 1) * BK;
      __builtin_amdgcn_global_load_async_to_lds_b128(
          (as1_v4i*)(Ag0 + ko), (as3_v4i*)&As[cur ^ 1][lr][lc], 0, 0);
      __builtin_amdgcn_global_load_async_to_lds_b128(
          (as1_v4i*)(Ag1 + ko), (as3_v4i*)&As[cur ^ 1][lr + 64][lc], 0, 0);
      __builtin_amdgcn_global_load_async_to_lds_b128(
          (as1_v4i*)(Wg + ko), (as3_v4i*)&Ws[cur ^ 1][lr][lc], 0, 0);
    }
#else
    float4 ra0, ra1, rw;
    if (k + 1 < nk) {
      const int ko = (k + 1) * BK;
      ra0 = *(const float4*)(Ag0 + ko);
      ra1 = *(const float4*)(Ag1 + ko);
      rw  = *(const float4*)(Wg + ko);
    }
#endif
#pragma unroll
    for (int kk = 0; kk < BK; kk += 4) {
      const int kr = kk + khalf;
      v2f a0, a1, b0, b1;
      a0.x = As[cur][wm + l15][kr];      a0.y = As[cur][wm + l15][kr + 1];
      a1.x = As[cur][wm + 16 + l15][kr]; a1.y = As[cur][wm + 16 + l15][kr + 1];
      b0.x = Ws[cur][wn + l15][kr];      b0.y = Ws[cur][wn + l15][kr + 1];
      b1.x = Ws[cur][wn + 16 + l15][kr]; b1.y = Ws[cur][wn + 16 + l15][kr + 1];
      acc00 = __builtin_amdgcn_wmma_f32_16x16x4_f32(false, a0, false, b0,
                                                    (short)0, acc00, false, false);
      acc01 = __builtin_amdgcn_wmma_f32_16x16x4_f32(false, a0, false, b1,
                                                    (short)0, acc01, false, false);
      acc10 = __builtin_amdgcn_wmma_f32_16x16x4_f32(false, a1, false, b0,
                                                    (short)0, acc10, false, false);
      acc11 = __builtin_amdgcn_wmma_f32_16x16x4_f32(false, a1, false, b1,
                                                    (short)0, acc11, false, false);
    }
#if USE_ASYNC_LDS
    if (k + 1 < nk) {
      __builtin_amdgcn_s_wait_asynccnt(0);
      __syncthreads();
    }
#else
    if (k + 1 < nk) {
      *(float4*)&As[cur ^ 1][lr][lc]      = ra0;
      *(float4*)&As[cur ^ 1][lr + 64][lc] = ra1;
      *(float4*)&Ws[cur ^ 1][lr][lc]      = rw;
      __syncthreads();
    }
#endif
  }

  const int n0 = bn + wn + l15;
  const float bi0 = bias[n0];
  const float bi1 = bias[n0 + 16];
#pragma unroll
  for (int r = 0; r < 8; ++r) {
    const int m0r = bm + wm + r + ((lane < 16) ? 0 : 8);
    const int m1r = m0r + 16;
    float v00 = acc00[r] + bi0;
    float v01 = acc01[r] + bi1;
    float v10 = acc10[r] + bi0;
    float v11 = acc11[r] + bi1;
    if (ACT == 1) {
      v00 = 1.0f / (1.0f + expf(-v00));
      v01 = 1.0f / (1.0f + expf(-v01));
      v10 = 1.0f / (1.0f + expf(-v10));
      v11 = 1.0f / (1.0f + expf(-v11));
    }
    size_t i00 = (size_t)m0r * Nc + n0, i01 = i00 + 16;
    size_t i10 = (size_t)m1r * Nc + n0, i11 = i10 + 16;
    if (EPI == 0) {
      C[i00] = v00; C[i01] = v01; C[i10] = v10; C[i11] = v11;
    } else {
      out2[i00] = v00; out2[i01] = v01; out2[i10] = v10; out2[i11] = v11;
      C[i00] = resid[i00] - v00; C[i01] = resid[i01] - v01;
      C[i10] = resid[i10] - v10; C[i11] = resid[i11] - v11;
    }
  }
}

// ---------------------------------------------------------------- FFT / freq attention
__device__ inline float4 insert4(float4 a, float v) {
  if (v > a.x)      { a.w = a.z; a.z = a.y; a.y = a.x; a.x = v; }
  else if (v > a.y) { a.w = a.z; a.z = a.y; a.y = v; }
  else if (v > a.z) { a.w = a.z; a.z = v; }
  else if (v > a.w) { a.w = v; }
  return a;
}

__device__ inline void fft2048(float2* sp, int t) {
  for (int j = 0; j < 8; ++j) {
    int n = t + 256 * j;
    int r = (int)(__brev((unsigned)n) >> 21);
    if (r > n) { float2 a = sp[n]; sp[n] = sp[r]; sp[r] = a; }
  }
  __syncthreads();
  for (int s = 0; s < 11; ++s) {
    int half = 1 << s;
#pragma unroll 1
    for (int j = 0; j < 4; ++j) {
      int idx = t + 256 * j;
      int grp = idx >> s;
      int pos = idx & (half - 1);
      int i0 = (grp << (s + 1)) + pos;
      int i1 = i0 + half;
      float ang = -(float)M_PI * (float)pos / (float)half;
      float sw, cw;
      __sincosf(ang, &sw, &cw);
      float2 u = sp[i0], v = sp[i1];
      float vr = v.x * cw - v.y * sw;
      float vi = v.x * sw + v.y * cw;
      sp[i0] = make_float2(u.x + vr, u.y + vi);
      sp[i1] = make_float2(u.x - vr, u.y - vi);
    }
    __syncthreads();
  }
}

// zt [B*D, N] in/out (z -= seasonal), lst [B*D, N] out (seasonal)
__global__ __launch_bounds__(256)
void k_freq_attn(float* __restrict__ zt, float* __restrict__ lst) {
  __shared__ float2 sp[N_];
  __shared__ float4 tops[256];
  const int row = blockIdx.x;
  const int t = threadIdx.x;
  float* zr = zt + (size_t)row * N_;
  float orig[8];
  for (int j = 0; j < 8; ++j) {
    int n = t + 256 * j;
    float v = zr[n];
    orig[j] = v;
    sp[n] = make_float2(v, 0.0f);
  }
  __syncthreads();
  fft2048(sp, t);

  float4 top = make_float4(-1.f, -1.f, -1.f, -1.f);
  for (int f = t; f <= 1024; f += 256) {
    float2 c = sp[f];
    top = insert4(top, sqrtf(c.x * c.x + c.y * c.y));
  }
  tops[t] = top;
  __syncthreads();
  for (int off = 128; off >= 1; off >>= 1) {
    if (t < off) {
      float4 o = tops[t + off];
      float4 m = tops[t];
      m = insert4(m, o.x); m = insert4(m, o.y);
      m = insert4(m, o.z); m = insert4(m, o.w);
      tops[t] = m;
    }
    __syncthreads();
  }
  float kth = tops[0].w;

  for (int j = 0; j < 8; ++j) {
    int f = t + 256 * j;
    float2 c = sp[f];
    float a = sqrtf(c.x * c.x + c.y * c.y);
    if (a < kth) { c.x = 0.0f; c.y = 0.0f; }
    c.y = -c.y;
    sp[f] = c;
  }
  __syncthreads();
  fft2048(sp, t);   // Re(ifft) = Re(fft(conj(X)))/N

  for (int j = 0; j < 8; ++j) {
    int n = t + 256 * j;
    float s = sp[n].x * (1.0f / (float)N_);
    lst[(size_t)row * N_ + n] = s;
    zr[n] = orig[j] - s;
  }
}

// ---------------------------------------------------------------- mhesa EMA scan
__global__ __launch_bounds__(512)
void k_mhesa_scan(const float* __restrict__ xin, float* __restrict__ outp,
                  const float* __restrict__ init_state, const float* __restrict__ alpha_raw) {
  int b = blockIdx.x;
  int d = threadIdx.x;
  int h = d >> 6;
  float a = 1.0f / (1.0f + expf(-alpha_raw[h]));
  float beta = 1.0f - a;
  float init = init_state[d];
  float prev = init, E = 0.0f, fac = 1.0f;
  const float* xr = xin + (size_t)b * N_ * D_ + d;
  float* orow = outp + (size_t)b * N_ * D_ + d;
  for (int n = 0; n < N_; ++n) {
    float x = xr[(size_t)n * D_];
    float diff = x - prev;
    prev = x;
    E = a * diff + beta * E;
    fac *= beta;
    orow[(size_t)n * D_] = E + fac * init;
  }
}

// ---------------------------------------------------------------- D -> TF projection (from [B,D,N])
__global__ __launch_bounds__(256)
void k_proj_tf(const float* __restrict__ lt, const float* __restrict__ W,
               const float* __restrict__ bias, float* __restrict__ out) {
  __shared__ float Ws[TF_ * D_];
  int b = blockIdx.y;
  int n = blockIdx.x * 256 + threadIdx.x;
  for (int i = threadIdx.x; i < TF_ * D_; i += 256) Ws[i] = W[i];
  __syncthreads();
  float acc[TF_];
#pragma unroll
  for (int f = 0; f < TF_; ++f) acc[f] = bias[f];
  const float* base = lt + (size_t)b * D_ * N_ + n;
  for (int d = 0; d < D_; ++d) {
    float v = base[(size_t)d * N_];
#pragma unroll
    for (int f = 0; f < TF_; ++f) acc[f] += v * Ws[f * D_ + d];
  }
  float* o = out + ((size_t)b * N_ + n) * TF_;
#pragma unroll
  for (int f = 0; f < TF_; ++f) o[f] = acc[f];
}

// ---------------------------------------------------------------- level scan
__global__ void k_level_scan(const float* __restrict__ x, const float* __restrict__ seas,
                             const float* __restrict__ grow, const float* __restrict__ alpha_raw,
                             float* __restrict__ xl_out) {
  int b = blockIdx.x;
  int f = threadIdx.x;
  if (f >= TF_) return;
  float a = 1.0f / (1.0f + expf(-alpha_raw[0]));
  float beta = 1.0f - a;
  float t1 = 0.0f, t2 = 0.0f;
  for (int n = 0; n < N_; ++n) {
    size_t i = ((size_t)b * N_ + n) * TF_ + f;
    t1 = a * (x[i] - seas[i]) + beta * t1;
    t2 = grow[i] + beta * t2;
    xl_out[i] = t1 + t2;
  }
}

// ---------------------------------------------------------------- summed accumulation
__global__ __launch_bounds__(256)
void k_accum(const float* __restrict__ lst, const float* __restrict__ lg,
             const float* __restrict__ dampen, float* __restrict__ summed) {
  int idx = blockIdx.x * 256 + threadIdx.x;
  if (idx >= B_ * T_ * D_) return;
  int d = idx & (D_ - 1);
  int tau = (idx >> 9) % T_;
  int b = idx / (T_ * D_);
  int h = d >> 6;
  float df = 1.0f / (1.0f + expf(-dampen[h]));
  float pw = df * (1.0f - powf(df, (float)(tau + 1))) / (1.0f - df);
  float lastg = lg[((size_t)b * N_ + (N_ - 1)) * D_ + d];
  float extra = lst[((size_t)b * D_ + d) * N_ + tau];
  summed[idx] += extra + lastg * pw;
}

// ---------------------------------------------------------------- layer norm
template <int RESID>
__global__ __launch_bounds__(256)
void k_layernorm(const float* __restrict__ x, const float* __restrict__ x2,
                 const float* __restrict__ g, const float* __restrict__ bta,
                 float* __restrict__ out) {
  __shared__ float sbuf[256];
  int row = blockIdx.x;
  int t = threadIdx.x;
  const float* xr = x + (size_t)row * D_;
  float v0 = xr[t], v1 = xr[t + 256];
  if (RESID) {
    const float* x2r = x2 + (size_t)row * D_;
    v0 += x2r[t]; v1 += x2r[t + 256];
  }
  sbuf[t] = v0 + v1;
  __syncthreads();
  for (int off = 128; off >= 1; off >>= 1) {
    if (t < off) sbuf[t] += sbuf[t + off];
    __syncthreads();
  }
  float mu = sbuf[0] * (1.0f / (float)D_);
  __syncthreads();
  float d0 = v0 - mu, d1 = v1 - mu;
  sbuf[t] = d0 * d0 + d1 * d1;
  __syncthreads();
  for (int off = 128; off >= 1; off >>= 1) {
    if (t < off) sbuf[t] += sbuf[t + off];
    __syncthreads();
  }
  float rs = rsqrtf(sbuf[0] * (1.0f / (float)D_) + 1e-5f);
  float* o = out + (size_t)row * D_;
  o[t]       = d0 * rs * g[t] + bta[t];
  o[t + 256] = d1 * rs * g[t + 256] + bta[t + 256];
}

// ---------------------------------------------------------------- final output
__global__ void k_final(const float* __restrict__ summed, const float* __restrict__ xl,
                        const float* __restrict__ ltf_w, const float* __restrict__ ltf_b,
                        float* __restrict__ out) {
  int idx = blockIdx.x * 128 + threadIdx.x;
  if (idx >= B_ * T_ * TF_) return;
  int f = idx % TF_;
  int tau = (idx / TF_) % T_;
  int b = idx / (TF_ * T_);
  float acc = ltf_b[f] + xl[((size_t)b * N_ + (N_ - 1)) * TF_ + f];
  const float* s = summed + ((size_t)b * T_ + tau) * D_;
  const float* wr = ltf_w + (size_t)f * D_;
  for (int d = 0; d < D_; ++d) acc += s[d] * wr[d];
  out[idx] = acc;
}

// ---------------------------------------------------------------- host
extern "C" void kernel_launch(void* const* d_in, const int* in_sizes, int n_in,
                              void* d_out, int out_size, void* d_ws, size_t ws_size,
                              hipStream_t stream) {
  const float* x        = (const float*)d_in[0];
  const float* embed_w  = (const float*)d_in[2];
  const float* embed_b  = (const float*)d_in[3];
  const float* mh_init  = (const float*)d_in[4];
  const float* mh_alpha = (const float*)d_in[5];
  const float* pin_w    = (const float*)d_in[6];
  const float* pin_b    = (const float*)d_in[7];
  const float* pout_w   = (const float*)d_in[8];
  const float* pout_b   = (const float*)d_in[9];
  const float* n1_g     = (const float*)d_in[10];
  const float* n1_b     = (const float*)d_in[11];
  const float* ff1_w    = (const float*)d_in[12];
  const float* ff1_b    = (const float*)d_in[13];
  const float* ff2_w    = (const float*)d_in[14];
  const float* ff2_b    = (const float*)d_in[15];
  const float* n2_g     = (const float*)d_in[16];
  const float* n2_b     = (const float*)d_in[17];
  const float* lvl_a    = (const float*)d_in[18];
  const float* tg_w     = (const float*)d_in[19];
  const float* tg_b     = (const float*)d_in[20];
  const float* ts_w     = (const float*)d_in[21];
  const float* ts_b     = (const float*)d_in[22];
  const float* dampen   = (const float*)d_in[23];
  const float* ltf_w    = (const float*)d_in[24];
  const float* ltf_b    = (const float*)d_in[25];
  float* out = (float*)d_out;

  float* ws = (float*)d_ws;
  const size_t SZ = (size_t)B_ * N_ * D_;
  const size_t SZ7 = (size_t)B_ * N_ * TF_;
  float* z     = ws;
  float* tmp1  = ws + SZ;
  float* tmp2  = ws + 2 * SZ;
  float* Sb    = ws + 3 * SZ;   // ls_t / ff output
  float* Gb    = ws + 4 * SZ;   // lg
  float* seas  = ws + 5 * SZ;
  float* grow  = seas + SZ7;
  float* xlA   = grow + SZ7;
  float* xlB   = xlA + SZ7;
  float* summed = xlB + SZ7;    // [B,T,D]

  dim3 tblk(32, 8);
  dim3 tg_nd(D_ / 32, N_ / 32, B_);   // [B,N,D] -> [B,D,N]
  dim3 tg_dn(N_ / 32, D_ / 32, B_);   // [B,D,N] -> [B,N,D]

  k_zero<<<(B_ * T_ * D_ + 255) / 256, 256, 0, stream>>>(summed, B_ * T_ * D_);
  k_embed<<<dim3(D_ / 256, N_, B_), 256, 0, stream>>>(x, embed_w, embed_b, z);

  for (int i = 0; i < 2; ++i) {
    // --- freq attention: ls = topK-freq(z); z -= ls ---
    k_transpose<<<tg_nd, tblk, 0, stream>>>(z, tmp1, N_, D_);
    k_freq_attn<<<B_ * D_, 256, 0, stream>>>(tmp1, Sb);
    k_transpose<<<tg_dn, tblk, 0, stream>>>(tmp1, z, D_, N_);

    // --- mhesa: lg = (EMA(diff(z@Win)))@Wout ; z -= lg ---
    k_gemm<0, 0><<<dim3(D_ / BN, BNROWS / BM), 256, 0, stream>>>(
        z, pin_w + (size_t)i * D_ * D_, pin_b + (size_t)i * D_, tmp1,
        nullptr, nullptr, BNROWS, D_, D_);
    k_mhesa_scan<<<B_, D_, 0, stream>>>(tmp1, tmp2,
        mh_init + (size_t)i * H_ * 64, mh_alpha + (size_t)i * H_);
    k_gemm<0, 1><<<dim3(D_ / BN, BNROWS / BM), 256, 0, stream>>>(
        tmp2, pout_w + (size_t)i * D_ * D_, pout_b + (size_t)i * D_, z,
        Gb, z, BNROWS, D_, D_);

    // --- level_fn (independent of FF -> run first to free Sb/Gb) ---
    k_transpose<<<tg_nd, tblk, 0, stream>>>(Gb, tmp1, N_, D_);  // lg_t
    k_proj_tf<<<dim3(N_ / 256, B_), 256, 0, stream>>>(
        Sb, ts_w + (size_t)i * TF_ * D_, ts_b + (size_t)i * TF_, seas);
    k_proj_tf<<<dim3(N_ / 256, B_), 256, 0, stream>>>(
        tmp1, tg_w + (size_t)i * TF_ * D_, tg_b + (size_t)i * TF_, grow);
    k_level_scan<<<B_, 32, 0, stream>>>(
        (i == 0) ? x : xlA, seas, grow, lvl_a + i, (i == 0) ? xlA : xlB);
    k_accum<<<(B_ * T_ * D_ + 255) / 256, 256, 0, stream>>>(Sb, Gb, dampen, summed);

    // --- feed-forward block (between layers only) ---
    if (i == 0) {
      k_layernorm<0><<<BNROWS, 256, 0, stream>>>(z, nullptr, n1_g, n1_b, tmp2);  // zn
      for (int c = 0; c < 4; ++c) {
        const size_t roff = (size_t)c * 8192 * D_;
        k_gemm<1, 0><<<dim3(D4_ / BN, 8192 / BM), 256, 0, stream>>>(
            tmp2 + roff, ff1_w, ff1_b, tmp1, nullptr, nullptr, 8192, D4_, D_);
        k_gemm<0, 0><<<dim3(D_ / BN, 8192 / BM), 256, 0, stream>>>(
            tmp1, ff2_w, ff2_b, Sb + roff, nullptr, nullptr, 8192, D_, D4_);
      }
      k_layernorm<1><<<BNROWS, 256, 0, stream>>>(tmp2, Sb, n2_g, n2_b, z);
    }
  }

  k_final<<<(B_ * T_ * TF_ + 127) / 128, 128, 0, stream>>>(summed, xlB, ltf_w, ltf_b, out);
}